// EdgeGATConv_56788057588255
// MI455X (gfx1250) — compile-verified
//
#include <hip/hip_runtime.h>
#include <hip/hip_bf16.h>

// CDNA5 / gfx1250, wave32. FP32 WMMA 16x16x4 for all dense projections.
typedef __attribute__((ext_vector_type(2))) float v2f;
typedef __attribute__((ext_vector_type(8))) float v8f;

#define NEG_SLOPE 0.2f
#define CDIM 128   // x feature dim
#define HDIM 16    // hidden dim
#define EDIM 16    // edge_attr dim
#define AWLD 48    // att_w row length = 2H + EDIM

// ---------------------------------------------------------------------------
// Kernel 1: h = x @ fc_w^T + fc_b      [N,128] x [128,16] -> [N,16]
// One wave per 16-row tile; K loop of 32 x (16x16x4) WMMAs.
// ---------------------------------------------------------------------------
__global__ void __launch_bounds__(256)
proj_h_kernel(const float* __restrict__ x, const float* __restrict__ fc_w,
              const float* __restrict__ fc_b, float* __restrict__ h, int N) {
    const int wave = threadIdx.x >> 5;
    const int lane = threadIdx.x & 31;
    const int tile = blockIdx.x * (blockDim.x >> 5) + wave;
    const int row0 = tile * 16;
    if (row0 >= N) return;                       // wave-uniform: EXEC stays full

    const int m  = lane & 15;                    // A row / B,C column
    const int kh = lane >> 4;                    // lane-half selects K pair

    int row = row0 + m;
    if (row >= N) row = N - 1;                   // clamp reads; stores masked
    const float* __restrict__ xrow = x + (size_t)row * CDIM;
    const float* __restrict__ wrow = fc_w + (size_t)m * CDIM;   // B[k][n]=fc_w[n][k]

    v8f c = {};
    #pragma unroll
    for (int k = 0; k < CDIM; k += 4) {
        v2f a, b;
        a.x = xrow[k + kh * 2 + 0]; a.y = xrow[k + kh * 2 + 1];
        b.x = wrow[k + kh * 2 + 0]; b.y = wrow[k + kh * 2 + 1];
        c = __builtin_amdgcn_wmma_f32_16x16x4_f32(false, a, false, b,
                                                  (short)0, c, false, false);
    }
    const float bias = fc_b[m];
    #pragma unroll
    for (int r = 0; r < 8; ++r) {                // D: row = r + kh*8, col = m
        const int rr = row0 + r + kh * 8;
        if (rr < N) h[(size_t)rr * HDIM + m] = c[r] + bias;
    }
}

// ---------------------------------------------------------------------------
// Kernel 2: per-node attention partials (L2-resident lookup tables)
//   as = h @ W1^T   (W1 = att_w[:, 0:16])
//   ad = h @ W2^T   (W2 = att_w[:,16:32])
// Shared A tile, two accumulators -> 8 WMMAs per 16 nodes.
// ---------------------------------------------------------------------------
__global__ void __launch_bounds__(256)
att_node_kernel(const float* __restrict__ h, const float* __restrict__ att_w,
                float* __restrict__ as_, float* __restrict__ ad_, int N) {
    const int wave = threadIdx.x >> 5;
    const int lane = threadIdx.x & 31;
    const int tile = blockIdx.x * (blockDim.x >> 5) + wave;
    const int row0 = tile * 16;
    if (row0 >= N) return;

    const int m  = lane & 15;
    const int kh = lane >> 4;
    int row = row0 + m;
    if (row >= N) row = N - 1;
    const float* __restrict__ hrow = h + (size_t)row * HDIM;
    const float* __restrict__ w    = att_w + (size_t)m * AWLD;  // row n of att_w

    v8f cs = {}, cd = {};
    #pragma unroll
    for (int k = 0; k < HDIM; k += 4) {
        v2f a, b1, b2;
        a.x  = hrow[k + kh * 2 + 0];        a.y  = hrow[k + kh * 2 + 1];
        b1.x = w[k + kh * 2 + 0];           b1.y = w[k + kh * 2 + 1];
        b2.x = w[HDIM + k + kh * 2 + 0];    b2.y = w[HDIM + k + kh * 2 + 1];
        cs = __builtin_amdgcn_wmma_f32_16x16x4_f32(false, a, false, b1,
                                                   (short)0, cs, false, false);
        cd = __builtin_amdgcn_wmma_f32_16x16x4_f32(false, a, false, b2,
                                                   (short)0, cd, false, false);
    }
    #pragma unroll
    for (int r = 0; r < 8; ++r) {
        const int rr = row0 + r + kh * 8;
        if (rr < N) {
            as_[(size_t)rr * HDIM + m] = cs[r];
            ad_[(size_t)rr * HDIM + m] = cd[r];
        }
    }
}

// ---------------------------------------------------------------------------
// Kernel 3 (streaming, dominates traffic): per 16-edge tile
//   t = edge_attr @ W3^T  (WMMA, W3 = att_w[:,32:48])
//   alpha = lrelu(t + as[src] + ad[dst] + att_b)
//   atomicAdd(out[src], h[dst] * alpha)      -> global_atomic_add_f32
// ---------------------------------------------------------------------------
__global__ void __launch_bounds__(256)
edge_kernel(const float* __restrict__ edge_attr,
            const long long* __restrict__ ei,       // [2,E] int64: src row, dst row
            const float* __restrict__ att_w, const float* __restrict__ att_b,
            const float* __restrict__ h, const float* __restrict__ as_,
            const float* __restrict__ ad_, float* __restrict__ out,
            long long E) {
    const int wave = threadIdx.x >> 5;
    const int lane = threadIdx.x & 31;
    const long long tile = (long long)blockIdx.x * (blockDim.x >> 5) + wave;
    const long long e0 = tile * 16;
    if (e0 >= E) return;

    const int m  = lane & 15;
    const int kh = lane >> 4;
    long long eA = e0 + m;
    if (eA >= E) eA = E - 1;
    const float* __restrict__ arow = edge_attr + eA * EDIM;
    const float* __restrict__ w    = att_w + (size_t)m * AWLD + 2 * HDIM;

    v8f c = {};
    #pragma unroll
    for (int k = 0; k < EDIM; k += 4) {
        v2f a, b;
        a.x = arow[k + kh * 2 + 0]; a.y = arow[k + kh * 2 + 1];
        b.x = w[k + kh * 2 + 0];    b.y = w[k + kh * 2 + 1];
        c = __builtin_amdgcn_wmma_f32_16x16x4_f32(false, a, false, b,
                                                  (short)0, c, false, false);
    }
    const float bias = att_b[m];
    const long long* __restrict__ srcp = ei;
    const long long* __restrict__ dstp = ei + E;
    #pragma unroll
    for (int r = 0; r < 8; ++r) {
        const long long e = e0 + r + kh * 8;     // D row -> edge id
        if (e >= E) continue;
        const long long s = srcp[e];
        const long long d = dstp[e];
        const float hj = h[d * HDIM + m];        // L2-resident gathers
        float t = c[r] + as_[s * HDIM + m] + ad_[d * HDIM + m] + bias;
        t = (t > 0.0f) ? t : NEG_SLOPE * t;      // leaky_relu
        atomicAdd(&out[s * HDIM + m], hj * t);   // segment_sum scatter-add
    }
}

// ---------------------------------------------------------------------------
extern "C" void kernel_launch(void* const* d_in, const int* in_sizes, int n_in,
                              void* d_out, int out_size, void* d_ws, size_t ws_size,
                              hipStream_t stream) {
    const float*     x         = (const float*)d_in[0];
    const long long* ei        = (const long long*)d_in[1];   // int64 per reference
    const float*     edge_attr = (const float*)d_in[2];
    const float*     fc_w      = (const float*)d_in[3];
    const float*     fc_b      = (const float*)d_in[4];
    const float*     att_w     = (const float*)d_in[5];
    const float*     att_b     = (const float*)d_in[6];
    float*           out       = (float*)d_out;

    const int       N = in_sizes[0] / CDIM;          // 50000
    const long long E = (long long)in_sizes[2] / EDIM; // 800000

    // workspace: h | as | ad   (3 * N * 16 floats = 9.6 MB)
    float* h   = (float*)d_ws;
    float* as_ = h   + (size_t)N * HDIM;
    float* ad_ = as_ + (size_t)N * HDIM;

    // out is accumulated via atomics -> zero it every call (capture-safe).
    hipMemsetAsync(out, 0, sizeof(float) * (size_t)out_size, stream);

    const int wavesPerBlock = 8;                 // 256 threads = 8 wave32
    const int tilesN  = (N + 15) / 16;
    const int blocksN = (tilesN + wavesPerBlock - 1) / wavesPerBlock;
    proj_h_kernel<<<blocksN, 256, 0, stream>>>(x, fc_w, fc_b, h, N);
    att_node_kernel<<<blocksN, 256, 0, stream>>>(h, att_w, as_, ad_, N);

    const long long tilesE = (E + 15) / 16;
    const int blocksE = (int)((tilesE + wavesPerBlock - 1) / wavesPerBlock);
    edge_kernel<<<blocksE, 256, 0, stream>>>(edge_attr, ei, att_w, att_b,
                                             h, as_, ad_, out, E);
}